// MyCRF_58909771432896
// MI455X (gfx1250) — compile-verified
//
#include <hip/hip_runtime.h>

#define SMALLV (-1000.0f)

constexpr int BB = 256;
constexpr int TT = 2048;
constexpr int KK = 64;
constexpr int KP = 65;    // augmented tag space
constexpr int CH = 64;    // emission rows per TDM chunk
constexpr int NCH = TT / CH;

typedef int v4i __attribute__((ext_vector_type(4)));
typedef int v8i __attribute__((ext_vector_type(8)));

// Issue a 1-D TDM copy: ndw dwords from global gptr -> LDS byte offset ldsOff.
// Wave-level op (EXEC ignored). Tracked by TENSORcnt.
__device__ __forceinline__ void tdm_load_1d(const void* gptr, unsigned ldsOff, unsigned ndw) {
    unsigned long long ga = (unsigned long long)gptr;
    v4i g0;
    g0.x = 1;                                   // count=1 (valid user descriptor)
    g0.y = (int)ldsOff;                         // lds_addr (bytes)
    g0.z = (int)(unsigned)ga;                   // global_addr[31:0]
    g0.w = (int)((((unsigned)(ga >> 32)) & 0x01FFFFFFu) | 0x80000000u); // addr[56:32] | type=2
    v8i g1;
    g1.s0 = 0x00020000;                         // wg_mask=0, data_size=4B
    g1.s1 = (int)(ndw << 16);                   // [63:48] tensor_dim0 lo16 ([47:32] atomic_addr=0)
    g1.s2 = (int)((ndw >> 16) | (1u << 16));    // [79:64] dim0 hi16, [95:80] tensor_dim1=1
    g1.s3 = (int)(ndw << 16);                   // [111:96] dim1 hi=0, [127:112] tile_dim0
    g1.s4 = 1;                                  // [143:128] tile_dim1=1, [159:144] tile_dim2=0
    g1.s5 = (int)ndw;                           // tensor_dim0_stride lo32
    g1.s6 = 0;                                  // stride0 hi16, stride1 lo16
    g1.s7 = 0;
    v4i gz; gz.x = 0; gz.y = 0; gz.z = 0; gz.w = 0; // groups 2/3: unused dims (all zero)
    asm volatile("tensor_load_to_lds %0, %1, %2, %3"
                 :: "s"(g0), "s"(g1), "s"(gz), "s"(gz) : "memory");
}

// One block per batch. 96 threads = 3 wave32s. Lane c owns state cur=c (c<65).
__global__ __launch_bounds__(96) void crf_viterbi_kernel(
    const float* __restrict__ x,      // [B,T,K]
    const int*   __restrict__ mask,   // [B,T]
    const float* __restrict__ trans,  // [KP,KP]
    float*       __restrict__ out,    // [B*T] paths (as float) then [B] best_score
    unsigned char* __restrict__ bp)   // workspace: [B,T,KP] backpointers (u8)
{
    __shared__ __align__(16) float transLDS[KP * KP + 3];   // 16912 B
    __shared__ __align__(16) float ebuf[2][CH * KK];        // 2 x 16384 B emission chunks
    __shared__ __align__(16) int   maskLDS[TT];             // 8192 B
    __shared__ __align__(16) float abuf[2][68];             // alpha double buffer
    __shared__ int lastTagS;

    const int tid = threadIdx.x;
    const int b   = blockIdx.x;
    const int c   = tid;                       // my cur state (valid if c < KP)
    const int ccK = (c < KK) ? c : 0;
    const int ccP = (c < KP) ? c : 0;

    const float* xb = x + (size_t)b * TT * KK;
    const int*   mb = mask + (size_t)b * TT;
    const size_t bpBase = (size_t)b * TT * KP;

    // ---- Stage trans (16900B) + mask row (8192B) into LDS with async global->LDS copies.
    {
        #pragma unroll
        for (int it = 0; it < 11; ++it) {                    // 1056 b128 chunks of trans
            int chunk = it * 96 + tid;
            unsigned ldsOff = (unsigned)(size_t)(&transLDS[chunk * 4]);
            const float* g = trans + chunk * 4;
            asm volatile("global_load_async_to_lds_b128 %0, %1, off"
                         :: "v"(ldsOff), "v"(g) : "memory");
        }
        if (tid == 0) transLDS[KP * KP - 1] = trans[KP * KP - 1];
        #pragma unroll
        for (int it = 0; it < 6; ++it) {                     // 512 b128 chunks of mask row
            int chunk = it * 96 + tid;
            if (chunk < TT / 4) {
                unsigned ldsOff = (unsigned)(size_t)(&maskLDS[chunk * 4]);
                const int* g = mb + chunk * 4;
                asm volatile("global_load_async_to_lds_b128 %0, %1, off"
                             :: "v"(ldsOff), "v"(g) : "memory");
            }
        }
        asm volatile("s_wait_asynccnt 0" ::: "memory");
    }

    // ---- Kick TDM emission pipeline: chunk0 (wait) + chunk1 (in flight).
    const unsigned ebOff0 = (unsigned)(size_t)(&ebuf[0][0]);
    const unsigned ebOff1 = (unsigned)(size_t)(&ebuf[1][0]);
    if (tid < 32) {
        tdm_load_1d(xb,               ebOff0, CH * KK);
        __builtin_amdgcn_s_wait_tensorcnt(0);
        tdm_load_1d(xb + CH * KK,     ebOff1, CH * KK);
    }

    // ---- Init alpha0 = [SMALL]*K ++ [0]
    float myA = (c == KK) ? 0.0f : SMALLV;
    if (c < KP) abuf[0][c] = myA;
    if (tid < 3) { abuf[0][KP + tid] = SMALLV; abuf[1][KP + tid] = SMALLV; }
    __syncthreads();   // trans/mask/chunk0 visible to all waves

    // ---- Pull my trans column into registers: tR[p] = trans[p][c]
    float tR[KP];
    #pragma unroll
    for (int p = 0; p < KP; ++p) tR[p] = transLDS[p * KP + ccP];

    auto step = [&](int t, float e, int m) {
        // Broadcast-read all 65 alphas (uniform-address ds_load_b128 x17)
        float aR[68];
        const float4* src = (const float4*)abuf[t & 1];
        #pragma unroll
        for (int i = 0; i < 17; ++i) {
            float4 v = src[i];
            aR[4*i+0] = v.x; aR[4*i+1] = v.y; aR[4*i+2] = v.z; aR[4*i+3] = v.w;
        }
        // argmax over prev (strict > keeps first max, matching jnp.argmax)
        float best = aR[0] + tR[0];
        int bpv = 0;
        #pragma unroll
        for (int p = 1; p < KP; ++p) {
            float cand = aR[p] + tR[p];
            bool g = cand > best;
            best = g ? cand : best;
            bpv  = g ? p : bpv;
        }
        float newA = best + e;
        bool mm = (m != 0);
        myA = mm ? newA : myA;
        int bpO = mm ? bpv : c;
        if (c < KP) {
            abuf[(t + 1) & 1][c] = myA;
            bp[bpBase + (size_t)t * KP + c] = (unsigned char)bpO;
        }
        __syncthreads();
    };

    // ---- Main loop: NCH chunks of CH steps, TDM double-buffered emissions.
    for (int ch = 0; ch < NCH; ++ch) {
        const float* eb = ebuf[ch & 1];
        float eC = (c < KK) ? eb[ccK] : SMALLV;          // emission for first step of chunk
        #pragma unroll 2
        for (int tt = 0; tt < CH; ++tt) {
            int t = ch * CH + tt;
            float eN = (c < KK && tt + 1 < CH) ? eb[(tt + 1) * KK + ccK] : SMALLV;
            int m = maskLDS[t];
            step(t, eC, m);
            eC = eN;
        }
        // Boundary: next chunk must be complete; kick chunk ch+2 into this buffer.
        if (ch + 1 < NCH) {
            if (tid < 32) {
                __builtin_amdgcn_s_wait_tensorcnt(0);
                if (ch + 2 < NCH)
                    tdm_load_1d(xb + (size_t)(ch + 2) * CH * KK,
                                (ch & 1) ? ebOff1 : ebOff0, CH * KK);
            }
            __syncthreads();
        }
    }

    // ---- Final argmax over alphaT (in abuf[0] since TT is even); first-max tie rule.
    if (tid == 0) {
        float bs = abuf[0][0]; int lt = 0;
        #pragma unroll 1
        for (int i = 1; i < KP; ++i) {
            float v = abuf[0][i];
            if (v > bs) { bs = v; lt = i; }
        }
        out[(size_t)BB * TT + b] = bs;
        lastTagS = lt;
    }

    // Make all waves' bp stores visible before wave0 backtracks.
    asm volatile("s_wait_storecnt 0x0" ::: "memory");
    __threadfence();
    __syncthreads();

    // ---- Backtrack by wave 0: 8-deep row prefetch ring + ds_bpermute column select.
    if (tid < 32) {
        const unsigned char* bpb = bp + bpBase;
        float* outb = out + (size_t)b * TT;
        int tag = lastTagS;

        unsigned r0[8], r1[8], r2[8];
        #pragma unroll
        for (int i = 0; i < 8; ++i) {
            const unsigned char* row = bpb + (size_t)(TT - 1 - i) * KP;
            r0[i] = row[tid];
            r1[i] = row[tid + 32];
            r2[i] = (tid == 0) ? (unsigned)row[64] : 0u;
        }

        #pragma unroll 1
        for (int tb = TT - 1; tb >= 0; tb -= 8) {
            #pragma unroll
            for (int i = 0; i < 8; ++i) {
                int t = tb - i;
                unsigned sel = (tag < 32) ? r0[i] : ((tag < 64) ? r1[i] : r2[i]);
                int got = __builtin_amdgcn_ds_bpermute((tag & 31) << 2, (int)sel);
                if (tid == 0) outb[t] = (float)tag;
                tag = got & 0xFF;
                int tn = t - 8;
                if (tn >= 0) {
                    const unsigned char* row = bpb + (size_t)tn * KP;
                    r0[i] = row[tid];
                    r1[i] = row[tid + 32];
                    r2[i] = (tid == 0) ? (unsigned)row[64] : 0u;
                }
            }
        }
    }
}

extern "C" void kernel_launch(void* const* d_in, const int* in_sizes, int n_in,
                              void* d_out, int out_size, void* d_ws, size_t ws_size,
                              hipStream_t stream) {
    const float* x     = (const float*)d_in[0];   // [B,T,K] f32
    const int*   mask  = (const int*)d_in[1];     // [B,T]   i32
    const float* trans = (const float*)d_in[2];   // [K+1,K+1] f32
    float* out = (float*)d_out;                   // [B*T] paths + [B] scores
    unsigned char* bp = (unsigned char*)d_ws;     // needs B*T*65 = ~34MB

    crf_viterbi_kernel<<<dim3(BB), dim3(96), 0, stream>>>(x, mask, trans, out, bp);
}